// RobustContrastNormalization_47356309405802
// MI455X (gfx1250) — compile-verified
//
#include <hip/hip_runtime.h>
#include <stdint.h>

// Problem constants (fixed shapes from the reference: x = [64, 512, 512, 3] f32)
#define NB 64
#define HW 262144                    // 512*512 pixels per batch image
#define GROUPS_TOTAL (NB * HW / 4)   // 4,194,304 groups of 4 pixels
#define BINS 65536
#define EPSILON 0.01f

// native clang vector (usable with __builtin_nontemporal_*)
typedef float v4f __attribute__((ext_vector_type(4)));

// -------------------------------------------------------------------------
// CDNA5 async global->LDS copy (ASYNCcnt path).  GVS addressing:
//   global addr = SADDR(64) + VADDR(32) ; LDS dest = per-lane VGPR byte addr.
// Each lane moves 16 bytes into its own LDS slot (same-lane round trip, so
// no cross-wave barrier is needed before consuming).
// -------------------------------------------------------------------------
__device__ __forceinline__ void async_ld_b128(unsigned lds_byte_addr,
                                              unsigned vofs_bytes,
                                              const float* sbase) {
  asm volatile("global_load_async_to_lds_b128 %0, %1, %2"
               :: "v"(lds_byte_addr), "v"(vofs_bytes), "s"(sbase)
               : "memory");
}
#define WAIT_ASYNC_1() asm volatile("s_wait_asynccnt 0x1" ::: "memory")
#define WAIT_ASYNC_0() asm volatile("s_wait_asynccnt 0x0" ::: "memory")

// -------------------------------------------------------------------------
// K1: channel mean.  Each "group" = 4 pixels = 12 consecutive floats of x =
// 3 consecutive float4 loads (coalesced b128).  Means stored as one float4
// into xm (aliases d_out to avoid workspace pressure).
// -------------------------------------------------------------------------
__global__ __launch_bounds__(256) void rcn_mean_kernel(
    const float* __restrict__ x, float* __restrict__ xm) {
  const float4* __restrict__ x4 = (const float4*)x;
  float4* __restrict__ m4 = (float4*)xm;
  const unsigned T = gridDim.x * 256u;             // total threads = 1,048,576
  const unsigned t = blockIdx.x * 256u + threadIdx.x;
  const float inv3 = 1.0f / 3.0f;
#pragma unroll
  for (int k = 0; k < 4; ++k) {
    const unsigned g = t + (unsigned)k * T;        // group id, < GROUPS_TOTAL
    if (k < 3) {
      // gfx1250 global_prefetch_b8 for the next iteration's cachelines
      __builtin_prefetch((const void*)&x4[(size_t)(g + T) * 3u], 0, 1);
    }
    const float4 a = x4[(size_t)g * 3u + 0u];
    const float4 b = x4[(size_t)g * 3u + 1u];
    const float4 c = x4[(size_t)g * 3u + 2u];
    float4 m;
    m.x = (a.x + a.y + a.z) * inv3;
    m.y = (a.w + b.x + b.y) * inv3;
    m.z = (b.z + b.w + c.x) * inv3;
    m.w = (c.y + c.z + c.w) * inv3;
    m4[g] = m;
  }
}

// -------------------------------------------------------------------------
// K2: per-batch quantiles.  65,536-bin LDS histogram (256 KB) + double-
// buffered async global->LDS staging (2 x 16 KB) + workgroup scan.
// Total LDS ~= 299 KB of the WGP's 320 KB.  One block (1024 thr) per batch.
// -------------------------------------------------------------------------
__global__ __launch_bounds__(1024) void rcn_quantile_kernel(
    const float* __restrict__ xm, float* __restrict__ res) {
  __shared__ unsigned hist[BINS];      // 262,144 B
  __shared__ float4 stage[2][1024];    //  32,768 B (async staging)
  __shared__ unsigned part[1024];      //   4,096 B
  __shared__ float sval[2];            // lo, hi

  const unsigned tid = threadIdx.x;
  const unsigned batch = blockIdx.x;

  // zero histogram
#pragma unroll
  for (int i = 0; i < BINS / 1024; ++i) hist[tid + i * 1024] = 0u;
  __syncthreads();

  // --- histogram phase: 64 chunks of 1024 float4, streamed through LDS ---
  const float* gbase = xm + (size_t)batch * HW;
  const unsigned lds0 = (unsigned)(uintptr_t)&stage[0][tid];
  const unsigned lds1 = (unsigned)(uintptr_t)&stage[1][tid];
  const unsigned vofs = tid * 16u;     // byte offset of this lane's float4

  async_ld_b128(lds0, vofs, gbase);    // prologue: chunk 0 in flight
  for (int i = 0; i < 64; ++i) {
    const unsigned cur = (unsigned)i & 1u;
    if (i < 63) {
      // issue chunk i+1 into the other buffer (16,384 B per chunk)
      async_ld_b128(cur ? lds0 : lds1, vofs + (unsigned)(i + 1) * 16384u, gbase);
      WAIT_ASYNC_1();                  // chunk i done (in-order completion)
    } else {
      WAIT_ASYNC_0();
    }
    const float4 v = stage[cur][tid];  // ds_load_b128 from staged tile
    int b0 = (int)(v.x * 65536.0f);
    int b1 = (int)(v.y * 65536.0f);
    int b2 = (int)(v.z * 65536.0f);
    int b3 = (int)(v.w * 65536.0f);
    b0 = b0 < 0 ? 0 : (b0 > BINS - 1 ? BINS - 1 : b0);
    b1 = b1 < 0 ? 0 : (b1 > BINS - 1 ? BINS - 1 : b1);
    b2 = b2 < 0 ? 0 : (b2 > BINS - 1 ? BINS - 1 : b2);
    b3 = b3 < 0 ? 0 : (b3 > BINS - 1 ? BINS - 1 : b3);
    atomicAdd(&hist[b0], 1u);          // ds_add_u32
    atomicAdd(&hist[b1], 1u);
    atomicAdd(&hist[b2], 1u);
    atomicAdd(&hist[b3], 1u);
  }
  __syncthreads();

  // per-thread sum over 64 consecutive bins
  const unsigned base = tid * 64u;
  unsigned csum = 0u;
  for (int i = 0; i < 64; ++i) csum += hist[base + i];
  part[tid] = csum;
  __syncthreads();

  // Hillis-Steele inclusive scan over the 1024 chunk sums
  for (unsigned off = 1u; off < 1024u; off <<= 1) {
    const unsigned v = part[tid];
    const unsigned add = (tid >= off) ? part[tid - off] : 0u;
    __syncthreads();
    part[tid] = v + add;
    __syncthreads();
  }
  const unsigned excl = part[tid] - csum;   // exclusive prefix of my chunk

  // locate quantile targets:  t = q * (n - 1), linear interpolation
  const float t0 = (float)(0.10 * (double)(HW - 1));   // 26214.3
  const float t1 = (float)(0.90 * (double)(HW - 1));   // 235928.7
  float cum = (float)excl;
  for (int i = 0; i < 64; ++i) {
    const unsigned c = hist[base + i];
    if (c) {
      const float fc = (float)c;
      const float lo = cum, hi = cum + fc;
      if (t0 >= lo && t0 < hi)
        sval[0] = ((float)(base + i) + (t0 - lo + 0.5f) / fc) * (1.0f / 65536.0f);
      if (t1 >= lo && t1 < hi)
        sval[1] = ((float)(base + i) + (t1 - lo + 0.5f) / fc) * (1.0f / 65536.0f);
      cum = hi;
    }
  }
  __syncthreads();

  if (tid == 0) {
    const float lo = sval[0];
    const float hi = sval[1];
    const float rng = fmaxf(hi - lo, EPSILON);
    res[batch * 2 + 0] = lo;
    res[batch * 2 + 1] = 1.0f / rng;
  }
}

// -------------------------------------------------------------------------
// K3: in-place normalize on d_out:  out = clip((xm - lo) * inv, 0, 1)
// Non-temporal load (last use of xm) + non-temporal store (don't pollute L2).
// Uses native ext-vector v4f (HIP float4 is a class and is rejected by the
// nontemporal builtins).
// -------------------------------------------------------------------------
__global__ __launch_bounds__(256) void rcn_norm_kernel(
    float* data, const float* __restrict__ res) {
  v4f* d4 = (v4f*)data;
  const unsigned T = gridDim.x * 256u;             // 1,048,576 threads
  const unsigned t = blockIdx.x * 256u + threadIdx.x;
#pragma unroll
  for (int k = 0; k < 4; ++k) {
    const unsigned q = t + (unsigned)k * T;        // v4f index
    const unsigned batch = q >> 16;                // 65,536 v4f per batch
    const float lo = res[batch * 2 + 0];
    const float inv = res[batch * 2 + 1];
    v4f v = __builtin_nontemporal_load(&d4[q]);
    v4f r;
    r.x = fminf(fmaxf((v.x - lo) * inv, 0.0f), 1.0f);
    r.y = fminf(fmaxf((v.y - lo) * inv, 0.0f), 1.0f);
    r.z = fminf(fmaxf((v.z - lo) * inv, 0.0f), 1.0f);
    r.w = fminf(fmaxf((v.w - lo) * inv, 0.0f), 1.0f);
    __builtin_nontemporal_store(r, &d4[q]);
  }
}

// -------------------------------------------------------------------------
extern "C" void kernel_launch(void* const* d_in, const int* in_sizes, int n_in,
                              void* d_out, int out_size, void* d_ws, size_t ws_size,
                              hipStream_t stream) {
  (void)in_sizes; (void)n_in; (void)out_size; (void)ws_size;
  const float* x = (const float*)d_in[0];       // [64,512,512,3] f32
  float* xm = (float*)d_out;                    // reuse output buffer as mean scratch
  float* res = (float*)d_ws;                    // 64 * {lo, 1/rng} = 512 B

  // K1: means (4096 blocks * 256 thr * 4 groups = 4,194,304 groups exactly)
  rcn_mean_kernel<<<dim3(4096), dim3(256), 0, stream>>>(x, xm);
  // K2: one workgroup per batch image
  rcn_quantile_kernel<<<dim3(NB), dim3(1024), 0, stream>>>(xm, res);
  // K3: in-place normalize
  rcn_norm_kernel<<<dim3(4096), dim3(256), 0, stream>>>(xm, res);
}